// Set2SetPooling_84335977824819
// MI455X (gfx1250) — compile-verified
//
#include <hip/hip_runtime.h>

#define N_NODES 4096
#define D       256
#define D2      512
#define D4      1024

typedef __attribute__((ext_vector_type(2))) float v2f;
typedef __attribute__((ext_vector_type(8))) float v8f;

// workspace layout (floats)
#define GT_OFF   0                         // GT[k][i] : 256 x 4096 (k-major)
#define EB_OFF   (D * N_NODES)             // eb[i]    : 4096
#define WQT_OFF  (EB_OFF + N_NODES)        // WqT packed [k/4][j][4] : 256 x 1024
#define WRT_OFF  (WQT_OFF + D * D4)        // WrT packed [k/4][j][4] : 256 x 1024
#define BS_OFF   (WRT_OFF + D * D4)        // bsum[j]  : 1024

// ---------------------------------------------------------------------------
// Fold W_ih[:, :D] + W_hh -> WqT, W_ih[:, D:] -> WrT, b_ih + b_hh -> bsum.
// Packed layout [k/4][j][4]: scan-side thread j loads a float4 of 4
// consecutive k weights -> coalesced global_load_b128 across the wave.
// ---------------------------------------------------------------------------
__global__ __launch_bounds__(256) void prep_weights(const float* __restrict__ W_ih,
                                                    const float* __restrict__ W_hh,
                                                    const float* __restrict__ b_ih,
                                                    const float* __restrict__ b_hh,
                                                    float* __restrict__ ws) {
    int idx = blockIdx.x * blockDim.x + threadIdx.x;   // 0 .. 262143
    int j = idx & (D4 - 1);                            // gate row 0..1023
    int k = idx >> 10;                                 // input dim 0..255
    int pk = ((k >> 2) * D4 + j) * 4 + (k & 3);        // packed index
    ws[WQT_OFF + pk] = W_ih[j * D2 + k] + W_hh[j * D + k];
    ws[WRT_OFF + pk] = W_ih[j * D2 + D + k];
    if (idx < D4) ws[BS_OFF + idx] = b_ih[idx] + b_hh[idx];
}

// ---------------------------------------------------------------------------
// G = h @ W_a via V_WMMA_F32_16X16X4_F32 (fp32 tensor path), stored transposed
// as GT[k][i]. One wave computes one 16x16 tile of G; 64 WMMAs over K=256.
// ---------------------------------------------------------------------------
__global__ __launch_bounds__(256) void prep_G_wmma(const float* __restrict__ h,
                                                   const float* __restrict__ Wa,
                                                   float* __restrict__ ws) {
    int wave = (blockIdx.x * blockDim.x + threadIdx.x) >> 5;   // 0..4095 tiles
    int lane = threadIdx.x & 31;
    int half = lane >> 4;        // 0: K pair {0,1}, 1: K pair {2,3}
    int lm   = lane & 15;        // M (for A) / N (for B/D)
    int mbase = (wave >> 4) * 16;   // 256 M-tiles
    int nbase = (wave & 15) * 16;   // 16  N-tiles

    const float* arow = h + (mbase + lm) * D;   // A: row M=lm
    v8f acc = {};
    for (int j = 0; j < D; j += 4) {
        int ka = j + 2 * half;
        // A 16x4: lanes 0-15 hold K=j,j+1 ; lanes 16-31 hold K=j+2,j+3
        v2f a; a.x = arow[ka]; a.y = arow[ka + 1];
        // B 4x16 (K x N), mirrored layout: per-lane consecutive K pair, N=lm
        v2f b; b.x = Wa[ka * D + nbase + lm];
               b.y = Wa[(ka + 1) * D + nbase + lm];
        acc = __builtin_amdgcn_wmma_f32_16x16x4_f32(
            false, a, false, b, (short)0, acc, false, false);
    }
    // C/D 16x16 f32: VGPR v -> M = v (lanes 0-15) / v+8 (lanes 16-31), N = lm
    float* GT = ws + GT_OFF;
#pragma unroll
    for (int v = 0; v < 8; ++v) {
        int M = mbase + v + 8 * half;
        int Ncol = nbase + lm;
        GT[Ncol * N_NODES + M] = acc[v];   // contiguous in M -> b128 stores
    }
}

// eb[i] = h[i,:] . b_a  (one-time)
__global__ __launch_bounds__(256) void prep_eb(const float* __restrict__ h,
                                               const float* __restrict__ b_a,
                                               float* __restrict__ ws) {
    int i = blockIdx.x * blockDim.x + threadIdx.x;   // 0..4095
    const float4* hr = (const float4*)(h + i * D);
    const float4* ba = (const float4*)b_a;
    float acc = 0.f;
#pragma unroll 4
    for (int k = 0; k < D / 4; ++k) {
        float4 a = hr[k], b = ba[k];
        acc += a.x * b.x + a.y * b.y + a.z * b.z + a.w * b.w;
    }
    ws[EB_OFF + i] = acc;
}

// ---------------------------------------------------------------------------
// Persistent single-workgroup scan: 4096 sequential steps, 1024 threads
// (32 wave32 waves). Recurrent state in LDS; every global access is a
// wave-coalesced b128 from k-major precomputed arrays (L2-resident:
// ~10MB/step << 192MB L2). vmem issue count is the WGP bottleneck -> 4x
// fewer instructions than a b32 formulation.
// ---------------------------------------------------------------------------
__global__ __launch_bounds__(1024) void scan_kernel(const float* __restrict__ h,
                                                    const float* __restrict__ ws,
                                                    float* __restrict__ out) {
    __shared__ __align__(16) float q[D], r[D], c[D];
    __shared__ __align__(16) float gates[D4];
    __shared__ __align__(16) float w[N_NODES];   // softmax wts; reused as scratch
    __shared__ float red[32];

    const int tid  = threadIdx.x;
    const int lane = tid & 31;
    const int wid  = tid >> 5;

    const float4* GT4  = (const float4*)(ws + GT_OFF);    // row k: 1024 float4
    const float4* eb4  = (const float4*)(ws + EB_OFF);
    const float4* Wq4  = (const float4*)(ws + WQT_OFF);   // [k/4][j]
    const float4* Wr4  = (const float4*)(ws + WRT_OFF);
    const float*  bs   = ws + BS_OFF;
    const float4* q4   = (const float4*)q;
    const float4* r4   = (const float4*)r;
    float4*       w4   = (float4*)w;

    if (tid < D) { q[tid] = 0.f; r[tid] = 0.f; c[tid] = 0.f; }
    __syncthreads();

    for (int t = 0; t < N_NODES; ++t) {
        // ---- phase 1: gates[j] = bsum[j] + Wq(:,j).q + Wr(:,j).r ----------
        float g = bs[tid];
        for (int kb = 0; kb < D / 4; ++kb) {
            float4 a  = Wq4[kb * D4 + tid];   // coalesced b128
            float4 b  = Wr4[kb * D4 + tid];   // coalesced b128
            float4 qv = q4[kb];               // ds_load_b128 broadcast
            float4 rv = r4[kb];
            g += a.x * qv.x + a.y * qv.y + a.z * qv.z + a.w * qv.w;
            g += b.x * rv.x + b.y * rv.y + b.z * rv.z + b.w * rv.w;
        }
        gates[tid] = g;
        __syncthreads();

        // ---- phase 2: LSTM cell update (threads 0..255) -------------------
        if (tid < D) {
            float ig = 1.f / (1.f + __expf(-gates[tid]));
            float fg = 1.f / (1.f + __expf(-gates[D + tid]));
            float gg = tanhf(gates[2 * D + tid]);
            float og = 1.f / (1.f + __expf(-gates[3 * D + tid]));
            float cn = fg * c[tid] + ig * gg;
            c[tid] = cn;
            q[tid] = og * tanhf(cn);
        }
        __syncthreads();

        // ---- phase 3: e = G q + eb, thread owns rows 4*tid .. 4*tid+3 -----
        float4 e4 = eb4[tid];                 // coalesced b128
        for (int k = 0; k < D; ++k) {
            float  qk = q[k];
            float4 gc = GT4[k * (N_NODES / 4) + tid];   // coalesced b128
            e4.x += gc.x * qk; e4.y += gc.y * qk;
            e4.z += gc.z * qk; e4.w += gc.w * qk;
        }
        // block max (wave32 shuffle + LDS)
        float lmax = fmaxf(fmaxf(e4.x, e4.y), fmaxf(e4.z, e4.w));
        for (int o = 16; o > 0; o >>= 1) lmax = fmaxf(lmax, __shfl_xor(lmax, o, 32));
        if (lane == 0) red[wid] = lmax;
        __syncthreads();
        if (wid == 0) {
            float m = red[lane];
            for (int o = 16; o > 0; o >>= 1) m = fmaxf(m, __shfl_xor(m, o, 32));
            if (lane == 0) red[0] = m;
        }
        __syncthreads();
        const float gmax = red[0];

        float4 wv;
        wv.x = __expf(e4.x - gmax); wv.y = __expf(e4.y - gmax);
        wv.z = __expf(e4.z - gmax); wv.w = __expf(e4.w - gmax);
        w4[tid] = wv;                          // ds_store_b128, w[4*tid+..]
        float lsum = wv.x + wv.y + wv.z + wv.w;
        for (int o = 16; o > 0; o >>= 1) lsum += __shfl_xor(lsum, o, 32);
        __syncthreads();                       // gmax reads done before red reuse
        if (lane == 0) red[wid] = lsum;
        __syncthreads();
        if (wid == 0) {
            float s = red[lane];
            for (int o = 16; o > 0; o >>= 1) s += __shfl_xor(s, o, 32);
            if (lane == 0) red[0] = s;
        }
        __syncthreads();
        const float inv = 1.f / red[0];

        // ---- phase 4: r[d] = inv * sum_i w[i] h[i,d] ----------------------
        // thread owns float4 of d (64 groups) x 16 row-chunks of 256 rows
        const int dq = tid & 63;               // float4 group within D
        const int ch = tid >> 6;               // 0..15
        const float4* hb = (const float4*)h + (ch * 256) * (D / 4) + dq;
        float4 acc = {0.f, 0.f, 0.f, 0.f};
        for (int i = 0; i < 256; ++i) {
            float wi = w[ch * 256 + i];        // LDS broadcast (uniform in wave)
            float4 hv = hb[i * (D / 4)];       // coalesced b128
            acc.x += wi * hv.x; acc.y += wi * hv.y;
            acc.z += wi * hv.z; acc.w += wi * hv.w;
        }
        __syncthreads();                       // all w reads done before reuse
        w4[ch * 64 + dq] = acc;                // scratch[ch][d] = partial
        __syncthreads();
        if (tid < D) {
            float s = 0.f;
#pragma unroll
            for (int cc = 0; cc < 16; ++cc) s += w[cc * D + tid];
            r[tid] = s * inv;
        }
        __syncthreads();
    }

    // q_star = concat(q, r)
    if (tid < D2) out[tid] = (tid < D) ? q[tid] : r[tid - D];
}

extern "C" void kernel_launch(void* const* d_in, const int* in_sizes, int n_in,
                              void* d_out, int out_size, void* d_ws, size_t ws_size,
                              hipStream_t stream) {
    const float* h    = (const float*)d_in[0];
    const float* W_ih = (const float*)d_in[1];
    const float* W_hh = (const float*)d_in[2];
    const float* b_ih = (const float*)d_in[3];
    const float* b_hh = (const float*)d_in[4];
    const float* W_a  = (const float*)d_in[5];
    const float* b_a  = (const float*)d_in[6];
    float* ws  = (float*)d_ws;
    float* out = (float*)d_out;
    (void)in_sizes; (void)n_in; (void)out_size; (void)ws_size;

    prep_weights<<<1024, 256, 0, stream>>>(W_ih, W_hh, b_ih, b_hh, ws);
    prep_G_wmma <<<512,  256, 0, stream>>>(h, W_a, ws);   // 4096 waves, 1 tile each
    prep_eb     <<<16,   256, 0, stream>>>(h, b_a, ws);
    scan_kernel <<<1,   1024, 0, stream>>>(h, ws, out);
}